// SimpleHeteroGNN_41704132444584
// MI455X (gfx1250) — compile-verified
//
#include <hip/hip_runtime.h>
#include <stdint.h>
#include <stddef.h>

// ---------------------------------------------------------------------------
// SimpleHeteroGNN on MI455X (gfx1250):
//   - memory-bound COO SpMM (gather + atomic scatter), fp32
//   - edge metadata streamed to LDS via gfx1250 async global->LDS path
//     (global_load_async_to_lds_b32 + s_wait_asynccnt, ASYNCcnt-tracked)
//   - 16-lane row teams (wave32 = 2 teams/wave), float4 (b128) row gathers
//   - native fp32 L2-side atomics via inline asm global_atomic_add_f32
//     (avoids clang's CAS-loop expansion of atomicrmw fadd)
// ---------------------------------------------------------------------------

#define N_USERS 100000
#define N_ITEMS 50000
#define N_CATS  1000
#define DIM     64
#define NLAYERS 3
#define NE_UI   2000000
#define NE_UU   1000000
#define NE_IC   150000

#define AS1 __attribute__((address_space(1)))
#define AS3 __attribute__((address_space(3)))

__device__ __forceinline__ void async_g2l_b32(const void* g, void* l) {
#if __has_builtin(__builtin_amdgcn_global_load_async_to_lds_b32)
  // builtin signature (probe-confirmed): (AS1 int*, AS3 int*, imm offset, imm cpol)
  __builtin_amdgcn_global_load_async_to_lds_b32((AS1 int*)g, (AS3 int*)l, 0, 0);
#else
  *(volatile unsigned*)l = *(const unsigned*)g;   // fallback: plain copy
#endif
}

__device__ __forceinline__ void wait_async0() {
#if __has_builtin(__builtin_amdgcn_s_wait_asynccnt)
  __builtin_amdgcn_s_wait_asynccnt(0);
#else
  asm volatile("s_wait_asynccnt 0" ::: "memory");
#endif
}

// Native no-return fp32 atomic adds into one 16B-aligned float4 slot.
// Address computed once; component selected with the offset immediate.
__device__ __forceinline__ void atom_add_f32x4(float* p, float4 v) {
  asm volatile("global_atomic_add_f32 %0, %1, off"           : : "v"(p), "v"(v.x) : "memory");
  asm volatile("global_atomic_add_f32 %0, %1, off offset:4"  : : "v"(p), "v"(v.y) : "memory");
  asm volatile("global_atomic_add_f32 %0, %1, off offset:8"  : : "v"(p), "v"(v.z) : "memory");
  asm volatile("global_atomic_add_f32 %0, %1, off offset:12" : : "v"(p), "v"(v.w) : "memory");
}

// ---------------------------------------------------------------------------
// SpMM: out[rows[e], :] += vals[e] * Hin[cols[e], :]   (atomic segment-sum)
// block = (16,16): tx = float4 chunk of the 64-wide row, ty = edge team.
// Each block stages a 256-edge tile of (rows, cols, vals) into LDS with
// async global->LDS copies, then its 16 teams sweep the tile.
// ---------------------------------------------------------------------------
#define EDGE_TILE 256

__global__ void __launch_bounds__(256)
spmm_atomic_kernel(const int* __restrict__ rows, const int* __restrict__ cols,
                   const float* __restrict__ vals,
                   const float* __restrict__ Hin, float* __restrict__ Hout,
                   int n_edges)
{
  __shared__ int   s_rows[EDGE_TILE];
  __shared__ int   s_cols[EDGE_TILE];
  __shared__ float s_vals[EDGE_TILE];

  const int tx  = threadIdx.x;          // 0..15 : feature chunk (float4)
  const int ty  = threadIdx.y;          // 0..15 : team id
  const int tid = ty * 16 + tx;         // 0..255

  const int base = blockIdx.x * EDGE_TILE;
  if (base >= n_edges) return;
  const int rem    = n_edges - base;
  const int tile_n = rem < EDGE_TILE ? rem : EDGE_TILE;

  if (tid < tile_n) {
    async_g2l_b32(rows + base + tid, &s_rows[tid]);
    async_g2l_b32(cols + base + tid, &s_cols[tid]);
    async_g2l_b32(vals + base + tid, &s_vals[tid]);
  }
  wait_async0();
  __syncthreads();

  for (int e = ty; e < tile_n; e += 16) {
    const int   r = s_rows[e];
    const int   c = s_cols[e];
    const float v = s_vals[e];
    // 16 lanes * float4 = one contiguous 256B row gather (global_load_b128)
    const float4 h = ((const float4*)(Hin + (size_t)c * DIM))[tx];
    float4 o;
    o.x = v * h.x; o.y = v * h.y; o.z = v * h.z; o.w = v * h.w;
    atom_add_f32x4(Hout + (size_t)r * DIM + tx * 4, o);
  }
}

// ---------------------------------------------------------------------------
// Elementwise helpers (float4-vectorized; all row counts are multiples of 4)
// ---------------------------------------------------------------------------
__global__ void __launch_bounds__(256)
dup2_kernel(const float4* __restrict__ s, float4* __restrict__ a,
            float4* __restrict__ b, int n4)
{
  int i = blockIdx.x * 256 + threadIdx.x;
  if (i < n4) { float4 x = s[i]; a[i] = x; b[i] = x; }
}

__global__ void __launch_bounds__(256)
zero_kernel(float4* __restrict__ p, int n4)
{
  int i = blockIdx.x * 256 + threadIdx.x;
  if (i < n4) p[i] = make_float4(0.f, 0.f, 0.f, 0.f);
}

__global__ void __launch_bounds__(256)
accum_kernel(float4* __restrict__ d, const float4* __restrict__ s, int n4)
{
  int i = blockIdx.x * 256 + threadIdx.x;
  if (i < n4) {
    float4 a = d[i]; float4 b = s[i];
    a.x += b.x; a.y += b.y; a.z += b.z; a.w += b.w;
    d[i] = a;
  }
}

__global__ void __launch_bounds__(256)
scale_kernel(float4* __restrict__ d, float sc, int n4)
{
  int i = blockIdx.x * 256 + threadIdx.x;
  if (i < n4) {
    float4 a = d[i];
    a.x *= sc; a.y *= sc; a.z *= sc; a.w *= sc;
    d[i] = a;
  }
}

__global__ void __launch_bounds__(256)
copy_kernel(float4* __restrict__ d, const float4* __restrict__ s, int n4)
{
  int i = blockIdx.x * 256 + threadIdx.x;
  if (i < n4) d[i] = s[i];
}

// ---------------------------------------------------------------------------
static inline int cdiv(long long a, long long b) { return (int)((a + b - 1) / b); }

extern "C" void kernel_launch(void* const* d_in, const int* in_sizes, int n_in,
                              void* d_out, int out_size, void* d_ws, size_t ws_size,
                              hipStream_t stream)
{
  (void)in_sizes; (void)n_in; (void)out_size; (void)ws_size;

  const float* user_emb = (const float*)d_in[0];
  const float* item_emb = (const float*)d_in[1];
  const float* cat_emb  = (const float*)d_in[2];
  const float* ui_vals  = (const float*)d_in[3];
  const float* iu_vals  = (const float*)d_in[4];
  const float* uu_vals  = (const float*)d_in[5];
  const float* ic_vals  = (const float*)d_in[6];
  const float* ci_vals  = (const float*)d_in[7];
  const int* ui_rows = (const int*)d_in[8];
  const int* ui_cols = (const int*)d_in[9];
  const int* iu_rows = (const int*)d_in[10];
  const int* iu_cols = (const int*)d_in[11];
  const int* uu_rows = (const int*)d_in[12];
  const int* uu_cols = (const int*)d_in[13];
  const int* ic_rows = (const int*)d_in[14];
  const int* ic_cols = (const int*)d_in[15];
  const int* ci_rows = (const int*)d_in[16];
  const int* ci_cols = (const int*)d_in[17];

  const size_t SZ_U = (size_t)N_USERS * DIM;   // 6,400,000 floats
  const size_t SZ_I = (size_t)N_ITEMS * DIM;   // 3,200,000
  const size_t SZ_C = (size_t)N_CATS  * DIM;   //    64,000

  // workspace: double-buffered h_u / h_i / h_c  (77.3 MB total)
  float* ws = (float*)d_ws;
  float* h_u[2] = { ws,                         ws + SZ_U };
  float* h_i[2] = { ws + 2*SZ_U,                ws + 2*SZ_U + SZ_I };
  float* h_c[2] = { ws + 2*SZ_U + 2*SZ_I,       ws + 2*SZ_U + 2*SZ_I + SZ_C };

  // outputs: sum_u | sum_i | h_c   (accumulate sums in-place in d_out)
  float* sum_u = (float*)d_out;
  float* sum_i = sum_u + SZ_U;
  float* out_c = sum_i + SZ_I;

  const int T = 256;
  const int n4_u = (int)(SZ_U / 4), n4_i = (int)(SZ_I / 4), n4_c = (int)(SZ_C / 4);
  const dim3 spmm_blk(16, 16, 1);

  // ---- init: h_*[0] = emb ; sum_* = emb (layer-0 contribution) ----
  dup2_kernel<<<cdiv(n4_u, T), T, 0, stream>>>((const float4*)user_emb,
      (float4*)h_u[0], (float4*)sum_u, n4_u);
  dup2_kernel<<<cdiv(n4_i, T), T, 0, stream>>>((const float4*)item_emb,
      (float4*)h_i[0], (float4*)sum_i, n4_i);
  copy_kernel<<<cdiv(n4_c, T), T, 0, stream>>>((float4*)h_c[0],
      (const float4*)cat_emb, n4_c);

  // ---- propagation layers ----
  int cur = 0;
  for (int l = 0; l < NLAYERS; ++l) {
    const int nxt = 1 - cur;

    zero_kernel<<<cdiv(n4_u, T), T, 0, stream>>>((float4*)h_u[nxt], n4_u);
    zero_kernel<<<cdiv(n4_i, T), T, 0, stream>>>((float4*)h_i[nxt], n4_i);
    zero_kernel<<<cdiv(n4_c, T), T, 0, stream>>>((float4*)h_c[nxt], n4_c);

    // new_h_u = UI @ h_i + UU @ h_u
    spmm_atomic_kernel<<<cdiv(NE_UI, EDGE_TILE), spmm_blk, 0, stream>>>(
        ui_rows, ui_cols, ui_vals, h_i[cur], h_u[nxt], NE_UI);
    spmm_atomic_kernel<<<cdiv(NE_UU, EDGE_TILE), spmm_blk, 0, stream>>>(
        uu_rows, uu_cols, uu_vals, h_u[cur], h_u[nxt], NE_UU);
    // new_h_i = IU @ h_u + IC @ h_c
    spmm_atomic_kernel<<<cdiv(NE_UI, EDGE_TILE), spmm_blk, 0, stream>>>(
        iu_rows, iu_cols, iu_vals, h_u[cur], h_i[nxt], NE_UI);
    spmm_atomic_kernel<<<cdiv(NE_IC, EDGE_TILE), spmm_blk, 0, stream>>>(
        ic_rows, ic_cols, ic_vals, h_c[cur], h_i[nxt], NE_IC);
    // new_h_c = CI @ h_i
    spmm_atomic_kernel<<<cdiv(NE_IC, EDGE_TILE), spmm_blk, 0, stream>>>(
        ci_rows, ci_cols, ci_vals, h_i[cur], h_c[nxt], NE_IC);

    accum_kernel<<<cdiv(n4_u, T), T, 0, stream>>>((float4*)sum_u,
        (const float4*)h_u[nxt], n4_u);
    accum_kernel<<<cdiv(n4_i, T), T, 0, stream>>>((float4*)sum_i,
        (const float4*)h_i[nxt], n4_i);

    cur = nxt;
  }

  // ---- finalize: scale sums by 1/(L+1), emit final h_c ----
  const float inv = 1.0f / (float)(NLAYERS + 1);
  scale_kernel<<<cdiv(n4_u, T), T, 0, stream>>>((float4*)sum_u, inv, n4_u);
  scale_kernel<<<cdiv(n4_i, T), T, 0, stream>>>((float4*)sum_i, inv, n4_i);
  copy_kernel<<<cdiv(n4_c, T), T, 0, stream>>>((float4*)out_c,
      (const float4*)h_c[cur], n4_c);
}